// CRF_19550691132083
// MI455X (gfx1250) — compile-verified
//
#include <hip/hip_runtime.h>
#include <stdint.h>

// ---------------------------------------------------------------------------
// CDNA5 feature detection (compile-only loop: guarded so we always compile).
// ---------------------------------------------------------------------------
#ifndef __has_builtin
#define __has_builtin(x) 0
#endif

#if __has_builtin(__builtin_amdgcn_global_load_async_to_lds_b128)
#define HAVE_ASYNC_LDS 1
#else
#define HAVE_ASYNC_LDS 0
#endif

#if __has_builtin(__builtin_amdgcn_s_wait_asynccnt)
#define HAVE_WAIT_ASYNC 1
#else
#define HAVE_WAIT_ASYNC 0
#endif

#if __has_builtin(__builtin_amdgcn_tensor_load_to_lds) && \
    __has_builtin(__builtin_amdgcn_s_wait_tensorcnt)
#define HAVE_TDM 1
#else
#define HAVE_TDM 0
#endif

// therock-10.0 headers ship the TDM descriptor header and use the 6-arg
// builtin; ROCm 7.2 has no header and a 5-arg builtin.
#if defined(__has_include)
#if __has_include(<hip/amd_detail/amd_gfx1250_TDM.h>)
#define TDM_6ARG 1
#else
#define TDM_6ARG 0
#endif
#else
#define TDM_6ARG 0
#endif

#define AS1 __attribute__((address_space(1)))
#define AS3 __attribute__((address_space(3)))

typedef int v4i __attribute__((ext_vector_type(4)));
typedef unsigned int v4u __attribute__((ext_vector_type(4)));
typedef int v8i __attribute__((ext_vector_type(8)));

namespace {
constexpr int kB = 512;
constexpr int kT = 2048;
constexpr int kU = 32;          // == wave32: lane j owns tag j
constexpr int kChunk = 64;      // timesteps staged per LDS chunk (8 KB)
constexpr int kIssues = (kChunk * kU * 4) / (32 * 16);  // 16 b128 issues/chunk
}  // namespace

// ---- async global -> LDS staging of one emission chunk (8192 B) -----------
__device__ __forceinline__ void stage_chunk_async(const float* gsrc, float* ldst, int lane) {
#if HAVE_ASYNC_LDS
#pragma unroll
  for (int k = 0; k < kIssues; ++k) {
    // each lane moves 16 B; 32 lanes x 16 issues = 8192 B, tracked on ASYNCcnt
    AS1 v4i* g = (AS1 v4i*)(gsrc + (k * 128 + lane * 4));
    AS3 v4i* l = (AS3 v4i*)(ldst + (k * 128 + lane * 4));
    __builtin_amdgcn_global_load_async_to_lds_b128(g, l, 0, 0);
  }
#else
#pragma unroll
  for (int k = 0; k < kIssues; ++k) {
    const float4 v = *(const float4*)(gsrc + (k * 128 + lane * 4));
    *(float4*)(ldst + (k * 128 + lane * 4)) = v;
  }
#endif
}

#if HAVE_TDM
// ---- TDM: one TENSOR_LOAD_TO_LDS moves the whole 8 KB chunk ---------------
__device__ __forceinline__ void stage_chunk_tdm(const float* gsrc, float* ldst) {
  const unsigned long long ga = (unsigned long long)(size_t)gsrc;
  const unsigned int lds_off = (unsigned int)(size_t)(AS3 char*)ldst;
  // D# group0: count=1 | lds_addr | global_addr[56:0] | type=2
  v4u g0 = {1u, lds_off, (unsigned int)ga,
            (unsigned int)((ga >> 32) & 0x01FFFFFFull) | 0x80000000u};
  // D# group1: data_size=4B (code 2); tensor_dim0=2048, tensor_dim1=1,
  //            tile_dim0=2048, tile_dim1=1, tensor_dim0_stride=2048
  v8i g1 = {0x00020000,  // workgroup_mask=0, data_size=2
            0x08000000,  // tensor_dim0[15:0]=2048 in bits[31:16]
            0x00010000,  // tensor_dim1[15:0]=1 in bits[31:16]
            0x08000000,  // tile_dim0=2048 in bits[31:16]
            1,           // tile_dim1=1, tile_dim2=0
            2048,        // tensor_dim0_stride[31:0]
            0, 0};
  v4i g2 = {0, 0, 0, 0};
  v4i g3 = {0, 0, 0, 0};
#if TDM_6ARG
  v8i g4 = {0, 0, 0, 0, 0, 0, 0, 0};
  __builtin_amdgcn_tensor_load_to_lds(g0, g1, g2, g3, g4, 0);
#else
  __builtin_amdgcn_tensor_load_to_lds(g0, g1, g2, g3, 0);
#endif
}
#endif

__device__ __forceinline__ void restage_chunk(const float* gsrc, float* ldst, int lane) {
#if HAVE_TDM
  (void)lane;
  stage_chunk_tdm(gsrc, ldst);
#else
  stage_chunk_async(gsrc, ldst, lane);
#endif
}

template <int N>
__device__ __forceinline__ void wait_async_le() {
#if HAVE_WAIT_ASYNC
  __builtin_amdgcn_s_wait_asynccnt((short)N);
#elif HAVE_ASYNC_LDS
  asm volatile("s_wait_asynccnt %0" ::"i"(N) : "memory");
#endif
  __asm__ __volatile__("" ::: "memory");  // compiler ordering fence
}

__device__ __forceinline__ void wait_chunk_ready(int c) {
#if HAVE_TDM
  // chunks 0/1 staged on ASYNCcnt; steady-state chunks on TENSORcnt
  if (c == 0) {
    wait_async_le<kIssues>();
  } else if (c == 1) {
    wait_async_le<0>();
  } else {
    __builtin_amdgcn_s_wait_tensorcnt(1);  // TDM in-order: newest may stay pending
    __asm__ __volatile__("" ::: "memory");
  }
#else
  (void)c;
  wait_async_le<kIssues>();
#endif
}

__device__ __forceinline__ void wait_lds_reads_done() {
#if HAVE_ASYNC_LDS || HAVE_TDM
#if __has_builtin(__builtin_amdgcn_s_wait_dscnt)
  __builtin_amdgcn_s_wait_dscnt(0);
#else
  asm volatile("s_wait_dscnt 0x0" ::: "memory");
#endif
#endif
}

__device__ __forceinline__ float bcast_f32(float v, int srclane) {
  return __int_as_float(__builtin_amdgcn_readlane(__float_as_int(v), srclane));
}

// ---------------------------------------------------------------------------
// Viterbi: one wave (32 lanes) per batch element; lane j == tag j.
// ---------------------------------------------------------------------------
__global__ __launch_bounds__(32) void viterbi_kernel(const float* __restrict__ pot,
                                                     const unsigned char* __restrict__ mask,
                                                     const float* __restrict__ trans,
                                                     float* __restrict__ out_tags,
                                                     float* __restrict__ out_len) {
  __shared__ float s_pot[2][kChunk * kU];        // 16 KB double buffer
  __shared__ uint32_t s_bp[(kT / 4) * kU];       // 64 KB: bp(t,j) = byte t&3 of word (t>>2)*U+j
  __shared__ unsigned char s_tags[kT];           // 2 KB traced tags

  const int lane = threadIdx.x;
  const int b = blockIdx.x;
  const float* gpot = pot + (size_t)b * kT * kU;

  // prefetch first two chunks (ASYNCcnt path)
  stage_chunk_async(gpot + 0 * kChunk * kU, &s_pot[0][0], lane);
  stage_chunk_async(gpot + 1 * kChunk * kU, &s_pot[1][0], lane);

  // ---- sequence length = popcount of contiguous prefix mask (overlaps async)
  const unsigned char* mrow = mask + (size_t)b * kT;
  int cnt = 0;
#pragma unroll
  for (int k = 0; k < kT / 128; ++k) {
    uchar4 m = reinterpret_cast<const uchar4*>(mrow)[k * 32 + lane];
    cnt += (int)m.x + (int)m.y + (int)m.z + (int)m.w;
  }
  int len = 0;
#pragma unroll
  for (int i = 0; i < 32; ++i) len += __builtin_amdgcn_readlane(cnt, i);
  if (len < 1) len = 1;
  if (len > kT) len = kT;

  // ---- transition column for this lane: trc[i] = trans[i][lane]
  float trc[kU];
#pragma unroll
  for (int i = 0; i < kU; ++i) trc[i] = trans[i * kU + lane];

  const int nchunks = (len + kChunk - 1) / kChunk;
  float state = 0.0f;

  for (int c = 0; c < nchunks; ++c) {
    wait_chunk_ready(c);
    const float* pb = &s_pot[c & 1][0];
    const int t0 = c * kChunk;
    const int tend = (len - t0 < kChunk) ? (len - t0) : kChunk;
    int tt = 0;
    if (c == 0) {  // t == 0: state initialized from emissions, no transition
      state = pb[lane];
      tt = 1;
    }
    uint32_t packed = 0;
#pragma unroll 1
    for (; tt < tend; ++tt) {
      const int t = t0 + tt;
      const float e = pb[tt * kU + lane];
      // new_state[lane] = e + max_i(state[i] + trans[i][lane]).
      // 4 independent chains (depth 8) for ILP/dual-issue; 'cand > bv' keeps
      // the FIRST max within a chain (jnp.argmax semantics).
      float bv[4];
      int bi[4];
#pragma unroll
      for (int l = 0; l < 4; ++l) {
        bv[l] = bcast_f32(state, l) + trc[l];
        bi[l] = l;
      }
#pragma unroll
      for (int i = 4; i < kU; ++i) {
        const int l = i & 3;
        const float cand = bcast_f32(state, i) + trc[i];
        if (cand > bv[l]) {
          bv[l] = cand;
          bi[l] = i;
        }
      }
      // merge: on equal values the smaller index wins -> global first-max
      float best = bv[0];
      int bidx = bi[0];
#pragma unroll
      for (int l = 1; l < 4; ++l) {
        if (bv[l] > best || (bv[l] == best && bi[l] < bidx)) {
          best = bv[l];
          bidx = bi[l];
        }
      }
      state = e + best;
      packed |= ((uint32_t)bidx) << ((t & 3) * 8);
      if ((t & 3) == 3 || tt == tend - 1) {
        s_bp[(t >> 2) * kU + lane] = packed;
        packed = 0;
      }
    }
    wait_lds_reads_done();
    if (c + 2 < nchunks)  // restage the buffer we just finished with chunk c+2
      restage_chunk(gpot + (size_t)(c + 2) * kChunk * kU, &s_pot[c & 1][0], lane);
  }

  // ---- last_tag = first-max argmax over lanes (uniform computation)
  float bval = bcast_f32(state, 0);
  int btag = 0;
#pragma unroll
  for (int i = 1; i < 32; ++i) {
    const float v = bcast_f32(state, i);
    if (v > bval) {
      bval = v;
      btag = i;
    }
  }

  // ---- padding region carries last_tag (identity backpointers in reference)
  for (int t = len - 1 + lane; t < kT; t += 32) s_tags[t] = (unsigned char)btag;

  // ---- backward trace, all inside LDS (dependent ds_load chain)
  if (lane == 0) {
    int tag = btag;
    for (int t = len - 2; t >= 0; --t) {
      const uint32_t w = s_bp[((t + 1) >> 2) * kU + tag];
      tag = (int)((w >> (((t + 1) & 3) * 8)) & 31u);
      s_tags[t] = (unsigned char)tag;
    }
  }
  __syncthreads();

  // ---- coalesced write-out of tags as float
  float* orow = out_tags + (size_t)b * kT;
#pragma unroll 1
  for (int t = lane; t < kT; t += 32) orow[t] = (float)s_tags[t];
  if (lane == 0) out_len[b] = (float)len;
}

// ---------------------------------------------------------------------------
// float4 grid-stride copy (inputs passthrough + trans passthrough)
// ---------------------------------------------------------------------------
__global__ void copy_f4_kernel(const float4* __restrict__ src, float4* __restrict__ dst, int n) {
  int i = blockIdx.x * blockDim.x + threadIdx.x;
  const int stride = gridDim.x * blockDim.x;
  for (; i < n; i += stride) dst[i] = src[i];
}

// ---------------------------------------------------------------------------
extern "C" void kernel_launch(void* const* d_in, const int* in_sizes, int n_in,
                              void* d_out, int out_size, void* d_ws, size_t ws_size,
                              hipStream_t stream) {
  (void)in_sizes; (void)n_in; (void)d_ws; (void)ws_size; (void)out_size;

  const float* pot = (const float*)d_in[0];                   // [B,T,U] f32
  const unsigned char* mask = (const unsigned char*)d_in[1];  // [B,T] bool (1B)
  const float* trans = (const float*)d_in[2];                 // [U,U] f32

  // d_out: tuple flattened in return order, float32:
  //   tags [B,T] | inputs [B,T,U] | lengths [B] | trans [U,U]
  float* out = (float*)d_out;
  float* out_tags = out;
  float* out_inputs = out + (size_t)kB * kT;
  float* out_len = out_inputs + (size_t)kB * kT * kU;
  float* out_trans = out_len + kB;

  viterbi_kernel<<<kB, 32, 0, stream>>>(pot, mask, trans, out_tags, out_len);
  copy_f4_kernel<<<8192, 256, 0, stream>>>((const float4*)pot, (float4*)out_inputs,
                                           (kB * kT * kU) / 4);
  copy_f4_kernel<<<4, 256, 0, stream>>>((const float4*)trans, (float4*)out_trans,
                                        (kU * kU) / 4);
}